// GMMConv_38448547234482
// MI455X (gfx1250) — compile-verified
//
#include <hip/hip_runtime.h>

typedef __attribute__((ext_vector_type(2))) float v2f;
typedef __attribute__((ext_vector_type(8))) float v8f;

#define C_IN  32
#define KCO   192   // K * C_out
#define C_OUT 32
#define NK    6
#define ND    3

// ---------------------------------------------------------------------------
// Prep: edge-invariant Gaussian coefficients.  inv[k][d] = -0.5/(eps+sigma^2),
// plus a contiguous copy of mu.  One tiny block; runs once per launch.
// ---------------------------------------------------------------------------
__global__ void prep_kernel(const float* __restrict__ mu,
                            const float* __restrict__ sigma,
                            float* __restrict__ inv, float* __restrict__ muc) {
  int i = threadIdx.x;
  if (i < NK * ND) {
    float sg = sigma[i];
    inv[i] = -0.5f / (1e-15f + sg * sg);
    muc[i] = mu[i];
  }
}

// ---------------------------------------------------------------------------
// Kernel A: xg = x @ g   ([N,32] @ [32,192]) via V_WMMA_F32_16X16X4_F32.
// One wave32 per 16x16 output tile; 4 waves (4 col-tiles) per 128-thread block.
// ---------------------------------------------------------------------------
__global__ __launch_bounds__(128)
void xg_wmma_kernel(const float* __restrict__ x, const float* __restrict__ g,
                    float* __restrict__ xg, int N) {
  const int lane = threadIdx.x & 31;
  const int wave = threadIdx.x >> 5;
  const int m0 = blockIdx.x * 16;
  const int j  = blockIdx.y * 4 + wave;   // column tile 0..11
  const int hx  = lane >> 4;              // half: 0 = lanes 0-15, 1 = lanes 16-31
  const int l16 = lane & 15;
  const int n0 = j * 16;

  int row = m0 + l16;
  if (row >= N) row = N - 1;              // clamp loads for the partial last tile

  v8f acc = {};
#pragma unroll
  for (int k = 0; k < C_IN; k += 4) {
    // A (16x4): lane holds M=l16, K = k + 2*hx + {0,1} -> contiguous pair
    v2f a = *(const v2f*)(x + row * C_IN + k + 2 * hx);
    // B (4x16): lane holds N=l16, K = k + 2*hx + {0,1}
    v2f b;
    b.x = g[(k + 2 * hx) * KCO + n0 + l16];
    b.y = g[(k + 2 * hx + 1) * KCO + n0 + l16];
    acc = __builtin_amdgcn_wmma_f32_16x16x4_f32(
        /*neg_a=*/false, a, /*neg_b=*/false, b,
        /*c_mod=*/(short)0, acc, /*reuse_a=*/false, /*reuse_b=*/false);
  }

  // D: vgpr i -> row m0 + i + 8*hx, col n0 + l16.
  // Uniform branch: only the last row-tile is partial.
  float* p = xg + (m0 + 8 * hx) * KCO + n0 + l16;
  if (m0 + 16 <= N) {
#pragma unroll
    for (int i = 0; i < 8; ++i) p[i * KCO] = acc[i];
  } else {
#pragma unroll
    for (int i = 0; i < 8; ++i) {
      int r = m0 + i + 8 * hx;
      if (r < N) xg[r * KCO + n0 + l16] = acc[i];
    }
  }
}

// ---------------------------------------------------------------------------
// Kernel B: per-edge Gaussian-mixture message + scatter-add.
// One wave per edge (lane = output channel) -> src/dst/gauss are wave-uniform.
// xg rows live in L2 (126 MB < 192 MB global L2).
// ---------------------------------------------------------------------------
__global__ __launch_bounds__(256)
void edge_kernel(const float* __restrict__ xg, const int* __restrict__ edge_index,
                 const float* __restrict__ pseudo, const float* __restrict__ inv,
                 const float* __restrict__ muc,
                 float* __restrict__ out, float* __restrict__ cnt, int E) {
  const int c = threadIdx.x & 31;
  const int e = blockIdx.x * 8 + (threadIdx.x >> 5);
  if (e >= E) return;

  const int src = edge_index[e];
  const int dst = edge_index[E + e];
  const float* xrow = xg + src * KCO;
  __builtin_prefetch(xrow + c, 0, 0);     // global_prefetch_b8: warm the gather row

  const float p0 = pseudo[e * ND + 0];
  const float p1 = pseudo[e * ND + 1];
  const float p2 = pseudo[e * ND + 2];

  float msg = 0.0f;
#pragma unroll
  for (int k = 0; k < NK; ++k) {
    float d0 = p0 - muc[k * ND + 0];
    float d1 = p1 - muc[k * ND + 1];
    float d2 = p2 - muc[k * ND + 2];
    float s = d0 * d0 * inv[k * ND + 0];
    s = fmaf(d1 * d1, inv[k * ND + 1], s);
    s = fmaf(d2 * d2, inv[k * ND + 2], s);
    msg = fmaf(__expf(s), xrow[k * C_OUT + c], msg);
  }

  __hip_atomic_fetch_add(out + dst * C_OUT + c, msg,
                         __ATOMIC_RELAXED, __HIP_MEMORY_SCOPE_AGENT);
  if (c == 0)
    __hip_atomic_fetch_add(cnt + dst, 1.0f,
                           __ATOMIC_RELAXED, __HIP_MEMORY_SCOPE_AGENT);
}

// ---------------------------------------------------------------------------
// Kernel C: out = out / max(cnt,1) + x @ root + bias, with x@root via WMMA.
// 2 waves per block (2 col-tiles of C_out=32); fused mean + bias in the
// WMMA C/D lane layout.
// ---------------------------------------------------------------------------
__global__ __launch_bounds__(64)
void finalize_kernel(const float* __restrict__ x, const float* __restrict__ root,
                     const float* __restrict__ bias, const float* __restrict__ cnt,
                     float* __restrict__ out, int N) {
  const int lane = threadIdx.x & 31;
  const int j = threadIdx.x >> 5;         // 0..1
  const int m0 = blockIdx.x * 16;
  const int hx  = lane >> 4;
  const int l16 = lane & 15;
  const int n0 = j * 16;

  int row = m0 + l16;
  if (row >= N) row = N - 1;

  v8f acc = {};
#pragma unroll
  for (int k = 0; k < C_IN; k += 4) {
    v2f a = *(const v2f*)(x + row * C_IN + k + 2 * hx);
    v2f b;
    b.x = root[(k + 2 * hx) * C_OUT + n0 + l16];
    b.y = root[(k + 2 * hx + 1) * C_OUT + n0 + l16];
    acc = __builtin_amdgcn_wmma_f32_16x16x4_f32(
        false, a, false, b, (short)0, acc, false, false);
  }

  const float bv = bias[n0 + l16];
  if (m0 + 16 <= N) {
    float* p = out + (m0 + 8 * hx) * C_OUT + n0 + l16;
    const float* cp = cnt + m0 + 8 * hx;
#pragma unroll
    for (int i = 0; i < 8; ++i) {
      float cv = cp[i];
      float denom = cv > 1.0f ? cv : 1.0f;
      p[i * C_OUT] = p[i * C_OUT] / denom + acc[i] + bv;
    }
  } else {
#pragma unroll
    for (int i = 0; i < 8; ++i) {
      int r = m0 + i + 8 * hx;
      if (r < N) {
        float cv = cnt[r];
        float denom = cv > 1.0f ? cv : 1.0f;
        int idx = r * C_OUT + n0 + l16;
        out[idx] = out[idx] / denom + acc[i] + bv;
      }
    }
  }
}

// ---------------------------------------------------------------------------
extern "C" void kernel_launch(void* const* d_in, const int* in_sizes, int n_in,
                              void* d_out, int out_size, void* d_ws, size_t ws_size,
                              hipStream_t stream) {
  const float* x          = (const float*)d_in[0];
  const int*   edge_index = (const int*)  d_in[1];
  const float* pseudo     = (const float*)d_in[2];
  const float* g          = (const float*)d_in[3];
  const float* mu         = (const float*)d_in[4];
  const float* sigma      = (const float*)d_in[5];
  const float* root       = (const float*)d_in[6];
  const float* bias       = (const float*)d_in[7];
  float* out = (float*)d_out;

  const int N = in_sizes[0] / C_IN;
  const int E = in_sizes[1] / 2;

  float* xg   = (float*)d_ws;                 // N * 192 floats (~126 MB, L2-resident)
  float* cnt  = xg + (size_t)N * KCO;         // N floats
  float* invc = cnt + N;                      // 18 floats
  float* muc  = invc + NK * ND;               // 18 floats

  hipMemsetAsync(out, 0, (size_t)out_size * sizeof(float), stream);
  hipMemsetAsync(cnt, 0, (size_t)N * sizeof(float), stream);

  prep_kernel<<<1, 32, 0, stream>>>(mu, sigma, invc, muc);

  const int mt = (N + 15) / 16;               // row tiles
  xg_wmma_kernel<<<dim3(mt, 3), 128, 0, stream>>>(x, g, xg, N);
  edge_kernel<<<(E + 7) / 8, 256, 0, stream>>>(xg, edge_index, pseudo, invc, muc,
                                               out, cnt, E);
  finalize_kernel<<<mt, 64, 0, stream>>>(x, root, bias, cnt, out, N);
}